// kmscm_38371237822847
// MI455X (gfx1250) — compile-verified
//
#include <hip/hip_runtime.h>
#include <math.h>

typedef __attribute__((ext_vector_type(2))) float v2f;
typedef __attribute__((ext_vector_type(8))) float v8f;
typedef __attribute__((address_space(3))) float lds_f;

#define NB    8
#define CIN   64
#define COUT  64
#define HWSZ  (384*384)        /* 147456 */
#define NK1   8
#define NK2   4
#define IT1   20
#define IT2   20
#define NPIX  128              /* pixels per conv block */
#define NBLK2 (HWSZ/256)       /* 576 */

/* ---------------- 1x1 conv as WMMA GEMM: y = w @ x + b ----------------
   x-tile staged to LDS with CDNA5 async-copy (GLOBAL_LOAD_ASYNC_TO_LDS_B128,
   tracked by ASYNCcnt), consumed by V_WMMA_F32_16X16X4_F32.               */
__global__ __launch_bounds__(256) void conv1x1_wmma(const float* __restrict__ x,
                                                    const float* __restrict__ w,
                                                    const float* __restrict__ bias,
                                                    float* __restrict__ y) {
  __shared__ __align__(16) float xt[CIN][NPIX + 4];   /* row stride 528B = 33*16B */
  const int b   = blockIdx.y;
  const int p0  = blockIdx.x * NPIX;
  const int tid = threadIdx.x;

  if (tid < COUT) __builtin_prefetch(&w[tid * CIN], 0, 0);

  /* async stage: 64 rows x 128 floats, 8 x B128 per lane, fully coalesced */
  const float* xb = x + (size_t)b * CIN * HWSZ + p0;
  const unsigned ldsbase = (unsigned)(uintptr_t)(lds_f*)&xt[0][0];
  #pragma unroll
  for (int it = 0; it < 8; ++it) {
    const int e4   = tid + it * 256;     /* vec4 index 0..2047 */
    const int c    = e4 >> 5;            /* row (channel) 0..63 */
    const int col4 = e4 & 31;            /* vec4 column within row */
    const unsigned voff = (unsigned)((c * HWSZ + col4 * 4) * 4);            /* global byte off */
    const unsigned ldst = ldsbase + (unsigned)((c * (NPIX + 4) + col4 * 4) * 4); /* LDS byte addr */
    asm volatile("global_load_async_to_lds_b128 %0, %1, %2"
                 :: "v"(ldst), "v"(voff), "s"(xb) : "memory");
  }
  asm volatile("s_wait_asynccnt 0x0" ::: "memory");
  __syncthreads();

  const int wave  = tid >> 5;
  const int lane  = tid & 31;
  const int half  = lane >> 4;       // 0: lanes 0-15, 1: lanes 16-31
  const int lan16 = lane & 15;
  const int m0    = (wave & 3) * 16; // output-channel tile base
  const int ngrp  = (wave >> 2) * 64;

  v8f acc[4];
  #pragma unroll
  for (int j = 0; j < 4; ++j) acc[j] = (v8f){};

  #pragma unroll 4
  for (int k = 0; k < 16; ++k) {
    const int c0 = 4 * k + 2 * half;       // K index base for this half-wave
    v2f a;
    a.x = w[(m0 + lan16) * CIN + c0];      // A[m][c0]
    a.y = w[(m0 + lan16) * CIN + c0 + 1];  // A[m][c0+1]
    #pragma unroll
    for (int j = 0; j < 4; ++j) {
      const int n = ngrp + 16 * j + lan16;
      v2f bf;
      bf.x = xt[c0][n];                    // B[c0][n]
      bf.y = xt[c0 + 1][n];                // B[c0+1][n]
      acc[j] = __builtin_amdgcn_wmma_f32_16x16x4_f32(
          false, a, false, bf, (short)0, acc[j], false, false);
    }
  }

  #pragma unroll
  for (int j = 0; j < 4; ++j) {
    const int pix = p0 + ngrp + 16 * j + lan16;
    #pragma unroll
    for (int r = 0; r < 8; ++r) {
      const int o = m0 + r + 8 * half;
      y[((size_t)b * COUT + o) * HWSZ + pix] = acc[j][r] + bias[o];
    }
  }
}

/* -------- stage 1: minmax-scale rows of y[0] (64 x HW) into xs -------- */
__global__ __launch_bounds__(256) void minmax_rows_s1(const float* __restrict__ y,
                                                      float* __restrict__ xs) {
  const int row = blockIdx.x;
  const float* p = y + (size_t)row * HWSZ;
  __shared__ float smn[256], smx[256];
  float mn = 3.4e38f, mx = -3.4e38f;
  for (int i = threadIdx.x; i < HWSZ; i += 256) {
    float v = p[i]; mn = fminf(mn, v); mx = fmaxf(mx, v);
  }
  smn[threadIdx.x] = mn; smx[threadIdx.x] = mx; __syncthreads();
  for (int s = 128; s > 0; s >>= 1) {
    if (threadIdx.x < s) {
      smn[threadIdx.x] = fminf(smn[threadIdx.x], smn[threadIdx.x + s]);
      smx[threadIdx.x] = fmaxf(smx[threadIdx.x], smx[threadIdx.x + s]);
    }
    __syncthreads();
  }
  mn = smn[0]; mx = smx[0];
  const float rng = mx - mn;
  const float den = (rng == 0.0f) ? 1.0f : rng;
  for (int i = threadIdx.x; i < HWSZ; i += 256) {
    float r = (p[i] - mn) / den;
    xs[(size_t)row * HWSZ + i] = isfinite(r) ? r : 0.0f;
  }
}

__global__ __launch_bounds__(256) void init_centers1(const float* __restrict__ xs,
                                                     float* __restrict__ c1) {
  const int k = blockIdx.y;
  const int hw = blockIdx.x * 256 + threadIdx.x;
  if (hw < HWSZ) c1[(size_t)k * HWSZ + hw] = xs[(size_t)(k * (COUT / NK1)) * HWSZ + hw];
}

__global__ __launch_bounds__(256) void cnorm1(const float* __restrict__ c1,
                                              float* __restrict__ cn1) {
  const int k = blockIdx.x;
  __shared__ float sm[256];
  float s = 0.0f;
  for (int i = threadIdx.x; i < HWSZ; i += 256) {
    float v = c1[(size_t)k * HWSZ + i]; s += v * v;
  }
  sm[threadIdx.x] = s; __syncthreads();
  for (int st = 128; st > 0; st >>= 1) {
    if (threadIdx.x < st) sm[threadIdx.x] += sm[threadIdx.x + st];
    __syncthreads();
  }
  if (threadIdx.x == 0) cn1[k] = sm[0];
}

__global__ __launch_bounds__(256) void assign1(const float* __restrict__ xs,
                                               const float* __restrict__ c1,
                                               const float* __restrict__ cn1,
                                               int* __restrict__ lab1,
                                               int* __restrict__ cnt1) {
  const int i = blockIdx.x;
  float part[NK1];
  #pragma unroll
  for (int k = 0; k < NK1; ++k) part[k] = 0.0f;
  for (int hw = threadIdx.x; hw < HWSZ; hw += 256) {
    const float p = xs[(size_t)i * HWSZ + hw];
    #pragma unroll
    for (int k = 0; k < NK1; ++k) part[k] += p * c1[(size_t)k * HWSZ + hw];
  }
  __shared__ float sm[NK1][256];
  #pragma unroll
  for (int k = 0; k < NK1; ++k) sm[k][threadIdx.x] = part[k];
  __syncthreads();
  for (int st = 128; st > 0; st >>= 1) {
    if (threadIdx.x < st) {
      #pragma unroll
      for (int k = 0; k < NK1; ++k) sm[k][threadIdx.x] += sm[k][threadIdx.x + st];
    }
    __syncthreads();
  }
  if (threadIdx.x == 0) {
    int best = 0;
    float bd = -2.0f * sm[0][0] + cn1[0];
    #pragma unroll
    for (int k = 1; k < NK1; ++k) {
      float d = -2.0f * sm[k][0] + cn1[k];
      if (d < bd) { bd = d; best = k; }
    }
    lab1[i] = best;
    atomicAdd(&cnt1[best], 1);
  }
}

__global__ __launch_bounds__(256) void update1(const float* __restrict__ xs,
                                               const float* __restrict__ c_old,
                                               const int* __restrict__ lab1,
                                               const int* __restrict__ cnt1,
                                               float* __restrict__ c_new) {
  const int k = blockIdx.y;
  const int hw = blockIdx.x * 256 + threadIdx.x;
  __shared__ int slab[COUT];
  __shared__ int scnt;
  if (threadIdx.x < COUT) slab[threadIdx.x] = lab1[threadIdx.x];
  if (threadIdx.x == 0) scnt = cnt1[k];
  __syncthreads();
  if (hw >= HWSZ) return;
  float s = 0.0f;
  for (int i = 0; i < COUT; ++i)
    if (slab[i] == k) s += xs[(size_t)i * HWSZ + hw];
  c_new[(size_t)k * HWSZ + hw] =
      (scnt > 0) ? s / fmaxf((float)scnt, 1.0f) : c_old[(size_t)k * HWSZ + hw];
}

/* -------- stage 2: per-image features (HW x 8), minmax per row -------- */
__global__ __launch_bounds__(256) void feat_kernel(const float* __restrict__ y,
                                                   const int* __restrict__ lab1,
                                                   const int* __restrict__ cnt1,
                                                   float* __restrict__ feat, int b) {
  __shared__ int slab[COUT];
  __shared__ float scc[NK1];
  if (threadIdx.x < COUT) slab[threadIdx.x] = lab1[threadIdx.x];
  if (threadIdx.x < NK1) scc[threadIdx.x] = fmaxf((float)cnt1[threadIdx.x], 1.0f);
  __syncthreads();
  const int hw = blockIdx.x * 256 + threadIdx.x;
  if (hw >= HWSZ) return;
  float acc[NK1];
  #pragma unroll
  for (int k = 0; k < NK1; ++k) acc[k] = 0.0f;
  const float* yb = y + (size_t)b * COUT * HWSZ;
  for (int c = 0; c < COUT; ++c) acc[slab[c]] += yb[(size_t)c * HWSZ + hw];
  float f[NK1], mn = 3.4e38f, mx = -3.4e38f;
  #pragma unroll
  for (int k = 0; k < NK1; ++k) {
    f[k] = acc[k] / scc[k];
    mn = fminf(mn, f[k]); mx = fmaxf(mx, f[k]);
  }
  const float rng = mx - mn;
  const float den = (rng == 0.0f) ? 1.0f : rng;
  #pragma unroll
  for (int k = 0; k < NK1; ++k) {
    float r = (f[k] - mn) / den;
    feat[(size_t)hw * NK1 + k] = isfinite(r) ? r : 0.0f;
  }
}

__global__ void copy32(const float* __restrict__ src, float* __restrict__ dst) {
  if (threadIdx.x < NK2 * NK1) dst[threadIdx.x] = src[threadIdx.x];
}

/* assign + deterministic per-block partial sums (no float atomics) */
__global__ __launch_bounds__(256) void assign2(const float* __restrict__ feat,
                                               const float* __restrict__ c2,
                                               int* __restrict__ lab2,
                                               int* __restrict__ cnt2,
                                               float* __restrict__ partial) {
  __shared__ float sc[NK2 * NK1];
  __shared__ float sf[256][NK1];
  __shared__ int slb[256];
  if (threadIdx.x < NK2 * NK1) sc[threadIdx.x] = c2[threadIdx.x];
  __syncthreads();
  float cn[NK2];
  #pragma unroll
  for (int k = 0; k < NK2; ++k) {
    float s = 0.0f;
    #pragma unroll
    for (int d = 0; d < NK1; ++d) s += sc[k * NK1 + d] * sc[k * NK1 + d];
    cn[k] = s;
  }
  const int hw = blockIdx.x * 256 + threadIdx.x;
  int lab = -1;
  if (hw < HWSZ) {
    float f[NK1];
    #pragma unroll
    for (int d = 0; d < NK1; ++d) f[d] = feat[(size_t)hw * NK1 + d];
    float bd = 3.4e38f;
    #pragma unroll
    for (int k = 0; k < NK2; ++k) {
      float dot = 0.0f;
      #pragma unroll
      for (int d = 0; d < NK1; ++d) dot += f[d] * sc[k * NK1 + d];
      float dd = -2.0f * dot + cn[k];
      if (dd < bd) { bd = dd; lab = k; }
    }
    lab2[hw] = lab;
    #pragma unroll
    for (int d = 0; d < NK1; ++d) sf[threadIdx.x][d] = f[d];
  }
  slb[threadIdx.x] = lab;
  __syncthreads();
  if (threadIdx.x < NK2 * NK1) {        // fixed-order reduction -> deterministic
    const int k = threadIdx.x >> 3, d = threadIdx.x & 7;
    float s = 0.0f;
    for (int i = 0; i < 256; ++i)
      if (slb[i] == k) s += sf[i][d];
    partial[(size_t)blockIdx.x * 32 + threadIdx.x] = s;
  }
  if (threadIdx.x < NK2) {
    int c = 0;
    for (int i = 0; i < 256; ++i) c += (slb[i] == (int)threadIdx.x);
    atomicAdd(&cnt2[threadIdx.x], c);   // integer atomics: deterministic
  }
}

__global__ void reduce2(const float* __restrict__ partial,
                        const int* __restrict__ cnt2, float* __restrict__ c2) {
  const int t = threadIdx.x;
  if (t >= NK2 * NK1) return;
  float s = 0.0f;
  for (int blk = 0; blk < NBLK2; ++blk) s += partial[(size_t)blk * 32 + t];
  const int c = cnt2[t >> 3];
  const float old = c2[t];
  c2[t] = (c > 0) ? s / fmaxf((float)c, 1.0f) : old;
}

/* replicate: mv=argsort(counts)[1]; swap 0<->mv; mes=((k)*frac)^2; stable argsort;
   mapvec; att = (mapped label == 0)                                            */
__global__ void post2(const int* __restrict__ cnt2, float* __restrict__ attlab) {
  int cnt[NK2];
  for (int k = 0; k < NK2; ++k) cnt[k] = cnt2[k];
  int idx[NK2] = {0, 1, 2, 3};
  for (int i = 1; i < NK2; ++i) {        // stable insertion sort ascending
    int key = idx[i], j = i - 1;
    while (j >= 0 && cnt[idx[j]] > cnt[key]) { idx[j + 1] = idx[j]; --j; }
    idx[j + 1] = key;
  }
  const int mv = idx[1];
  int cs[NK2];
  for (int k = 0; k < NK2; ++k) cs[k] = cnt[k];
  { int t = cs[0]; cs[0] = cs[mv]; cs[mv] = t; }
  float mes[3];
  for (int j = 0; j < 3; ++j) {
    float frac = (float)cs[j + 1] / (float)HWSZ;
    float m = (float)(j + 1) * frac;
    mes[j] = m * m;
  }
  int id2[3] = {0, 1, 2};
  for (int i = 1; i < 3; ++i) {
    int key = id2[i], j = i - 1;
    while (j >= 0 && mes[id2[j]] > mes[key]) { id2[j + 1] = id2[j]; --j; }
    id2[j + 1] = key;
  }
  int mapv[NK2] = {0, 1, 2, 3};
  mapv[id2[0] + 1] = 0; mapv[id2[1] + 1] = 1; mapv[id2[2] + 1] = 1;
  for (int j = 0; j < NK2; ++j) {
    int sw = (j == mv) ? 0 : ((j == 0) ? mv : j);
    attlab[j] = (mapv[sw] == 0) ? 1.0f : 0.0f;
  }
}

__global__ __launch_bounds__(256) void writeatt(const int* __restrict__ lab2,
                                                const float* __restrict__ attlab,
                                                float* __restrict__ attm) {
  const int hw = blockIdx.x * 256 + threadIdx.x;
  if (hw < HWSZ) attm[hw] = attlab[lab2[hw]];
}

__global__ __launch_bounds__(256) void final_scale(float* __restrict__ y,
                                                   const float* __restrict__ attm) {
  const size_t idx = (size_t)blockIdx.x * 256 + threadIdx.x;
  const size_t total = (size_t)NB * COUT * HWSZ;
  if (idx >= total) return;
  const int b = (int)(idx / ((size_t)COUT * HWSZ));
  const int hw = (int)(idx % HWSZ);
  const float a = attm[(size_t)b * HWSZ + hw];
  const float sg = 1.0f / (1.0f + expf(-a));
  y[idx] = y[idx] * (1.0f + sg);
}

extern "C" void kernel_launch(void* const* d_in, const int* in_sizes, int n_in,
                              void* d_out, int out_size, void* d_ws, size_t ws_size,
                              hipStream_t stream) {
  const float* x  = (const float*)d_in[0];   // (8,64,384,384)
  const float* w  = (const float*)d_in[1];   // (64,64)
  const float* bb = (const float*)d_in[2];   // (64,)
  const float* cc = (const float*)d_in[3];   // (4,8)
  float* y = (float*)d_out;                  // (8,64,384,384)

  /* workspace carve (~58 MB) */
  float* xs      = (float*)d_ws;                       // 64*HW
  float* c1a     = xs + (size_t)64 * HWSZ;             // 8*HW
  float* c1b     = c1a + (size_t)NK1 * HWSZ;           // 8*HW
  float* cn1     = c1b + (size_t)NK1 * HWSZ;           // 8
  float* feat    = cn1 + NK1;                          // HW*8
  float* c2      = feat + (size_t)HWSZ * NK1;          // 32
  float* partial = c2 + NK2 * NK1;                     // NBLK2*32
  float* attlab  = partial + (size_t)NBLK2 * 32;       // 4
  float* attm    = attlab + NK2;                       // NB*HW
  int* lab1 = (int*)(attm + (size_t)NB * HWSZ);        // 64
  int* cnt1 = lab1 + COUT;                             // 8
  int* lab2 = cnt1 + NK1;                              // HW
  int* cnt2 = lab2 + HWSZ;                             // 4

  /* 1) conv (WMMA + async tile staging) */
  conv1x1_wmma<<<dim3(HWSZ / NPIX, NB), 256, 0, stream>>>(x, w, bb, y);

  /* 2) stage-1 kmeans on channels of sample 0 */
  minmax_rows_s1<<<COUT, 256, 0, stream>>>(y, xs);
  init_centers1<<<dim3(NBLK2, NK1), 256, 0, stream>>>(xs, c1a);
  float* ca = c1a; float* cb = c1b;
  for (int it = 0; it < IT1; ++it) {
    cnorm1<<<NK1, 256, 0, stream>>>(ca, cn1);
    hipMemsetAsync(cnt1, 0, NK1 * sizeof(int), stream);
    assign1<<<COUT, 256, 0, stream>>>(xs, ca, cn1, lab1, cnt1);
    update1<<<dim3(NBLK2, NK1), 256, 0, stream>>>(xs, ca, lab1, cnt1, cb);
    float* t = ca; ca = cb; cb = t;
  }
  cnorm1<<<NK1, 256, 0, stream>>>(ca, cn1);
  hipMemsetAsync(cnt1, 0, NK1 * sizeof(int), stream);
  assign1<<<COUT, 256, 0, stream>>>(xs, ca, cn1, lab1, cnt1);

  /* 3) stage-2 per-image kmeans, centers carried across batch */
  copy32<<<1, 32, 0, stream>>>(cc, c2);
  for (int b = 0; b < NB; ++b) {
    feat_kernel<<<NBLK2, 256, 0, stream>>>(y, lab1, cnt1, feat, b);
    for (int it = 0; it < IT2; ++it) {
      hipMemsetAsync(cnt2, 0, NK2 * sizeof(int), stream);
      assign2<<<NBLK2, 256, 0, stream>>>(feat, c2, lab2, cnt2, partial);
      reduce2<<<1, 32, 0, stream>>>(partial, cnt2, c2);
    }
    hipMemsetAsync(cnt2, 0, NK2 * sizeof(int), stream);
    assign2<<<NBLK2, 256, 0, stream>>>(feat, c2, lab2, cnt2, partial);
    post2<<<1, 1, 0, stream>>>(cnt2, attlab);
    writeatt<<<NBLK2, 256, 0, stream>>>(lab2, attlab, attm + (size_t)b * HWSZ);
  }

  /* 4) out = y * (1 + sigmoid(att)) */
  const size_t total = (size_t)NB * COUT * HWSZ;
  final_scale<<<(unsigned)((total + 255) / 256), 256, 0, stream>>>(y, attm);
}